// DS_GCNPredictor_63969242907021
// MI455X (gfx1250) — compile-verified
//
#include <hip/hip_runtime.h>

typedef float v2f __attribute__((ext_vector_type(2)));
typedef float v8f __attribute__((ext_vector_type(8)));

#define N_NODES 100000
#define NE      1000000
#define NEL     500000
#define NSURF   3
#define FD      64          // feature dim for all hidden layers

// ---------------------------------------------------------------------------
// GEMM: Y[N,64] = (RELU ? relu(X) : X)[N,64] @ W[64,64] via V_WMMA_F32_16X16X4_F32.
// W staged in LDS once per block (8 waves share it). Each wave computes two
// 16-row tiles (32 rows) so each B fragment feeds 2 WMMAs: 128 WMMAs per wave.
// A-frag: lane(lo,half) holds A[lo][4*ks+2*half+j]; B-frag: B[4*ks+2*half+j][lo];
// C: c[r] = C[r+8*half][lo]  (per CDNA5 ISA 7.12.2 layouts).
// ---------------------------------------------------------------------------
template <int RELU>
__global__ void gemm64_wmma(const float* __restrict__ X,
                            const float* __restrict__ W,
                            float* __restrict__ Y, int n) {
    __shared__ float Ws[FD * FD];
    {   // cooperative 16 KB weight stage: 256 threads x 4 float4
        const float4* Wv = (const float4*)W;
        float4*       Sv = (float4*)Ws;
        for (int i = threadIdx.x; i < FD * FD / 4; i += blockDim.x)
            Sv[i] = Wv[i];
    }
    __syncthreads();

    const int wave = blockIdx.x * (blockDim.x >> 5) + (threadIdx.x >> 5);
    const int row0 = wave * 32;                // two 16-row tiles per wave
    if (row0 >= n) return;                     // wave-uniform: EXEC stays all-ones
    const int lane = threadIdx.x & 31;
    const int lo   = lane & 15;
    const int half = lane >> 4;
    const float* xr0 = X + (size_t)(row0 + lo) * FD;
    const float* xr1 = xr0 + (size_t)16 * FD;

    #pragma unroll
    for (int ct = 0; ct < 4; ++ct) {
        const int col = ct * 16 + lo;
        v8f acc0 = {0.f, 0.f, 0.f, 0.f, 0.f, 0.f, 0.f, 0.f};
        v8f acc1 = {0.f, 0.f, 0.f, 0.f, 0.f, 0.f, 0.f, 0.f};
        #pragma unroll
        for (int ks = 0; ks < 16; ++ks) {
            const int k0 = ks * 4 + half * 2;
            v2f a0, a1, b;
            if (RELU) {
                a0[0] = fmaxf(xr0[k0], 0.f); a0[1] = fmaxf(xr0[k0 + 1], 0.f);
                a1[0] = fmaxf(xr1[k0], 0.f); a1[1] = fmaxf(xr1[k0 + 1], 0.f);
            } else {
                a0[0] = xr0[k0]; a0[1] = xr0[k0 + 1];
                a1[0] = xr1[k0]; a1[1] = xr1[k0 + 1];
            }
            b[0] = Ws[k0 * FD + col];
            b[1] = Ws[(k0 + 1) * FD + col];
            acc0 = __builtin_amdgcn_wmma_f32_16x16x4_f32(
                       false, a0, false, b, (short)0, acc0, false, false);
            acc1 = __builtin_amdgcn_wmma_f32_16x16x4_f32(
                       false, a1, false, b, (short)0, acc1, false, false);
        }
        #pragma unroll
        for (int r = 0; r < 8; ++r) {
            Y[(size_t)(row0 + r + half * 8) * FD + col]      = acc0[r];
            Y[(size_t)(row0 + 16 + r + half * 8) * FD + col] = acc1[r];
        }
    }
}

// ---------------------------------------------------------------------------
// Degree / normalization helpers
// ---------------------------------------------------------------------------
__global__ void fill_kernel(float* __restrict__ p, float v, int n) {
    int t = blockIdx.x * blockDim.x + threadIdx.x;
    if (t < n) p[t] = v;
}

__global__ void deg_edges_kernel(const int* __restrict__ dst,
                                 float* __restrict__ deg, int ne) {
    int t = blockIdx.x * blockDim.x + threadIdx.x;
    if (t < ne) atomicAdd(&deg[dst[t]], 1.0f);
}

__global__ void dinv_kernel(const float* __restrict__ deg,
                            float* __restrict__ dinv, int n) {
    int t = blockIdx.x * blockDim.x + threadIdx.x;
    if (t < n) dinv[t] = rsqrtf(fmaxf(deg[t], 1.0f));
}

// ---------------------------------------------------------------------------
// GCN aggregation: out[n,k] = dinv[n]^2*h[n,k] + bias[k]  then += edge messages
// ---------------------------------------------------------------------------
template <int D>
__global__ void agg_self_kernel(const float* __restrict__ h,
                                const float* __restrict__ dinv,
                                const float* __restrict__ bias,
                                float* __restrict__ out, int n) {
    int t = blockIdx.x * blockDim.x + threadIdx.x;
    if (t >= n * D) return;
    int node = t / D, k = t % D;
    float di = dinv[node];
    out[t] = di * di * h[t] + bias[k];
}

template <int D>
__global__ void agg_edges_kernel(const float* __restrict__ h,
                                 const int* __restrict__ src,
                                 const int* __restrict__ dst,
                                 const float* __restrict__ dinv,
                                 float* __restrict__ out, int ne) {
    int t = blockIdx.x * blockDim.x + threadIdx.x;
    if (t >= ne * D) return;
    int e = t / D, k = t % D;
    int s = src[e], r = dst[e];
    atomicAdd(&out[(size_t)r * D + k], dinv[s] * dinv[r] * h[(size_t)s * D + k]);
}

// ---------------------------------------------------------------------------
// Cross-layer aggregation: out[recv[e],k] += last[src[e],k]; then scale by 1/deg
// ---------------------------------------------------------------------------
__global__ void cross_edges_kernel(const float* __restrict__ last,
                                   const int* __restrict__ recv,
                                   const int* __restrict__ src,
                                   float* __restrict__ out, int ne) {
    int t = blockIdx.x * blockDim.x + threadIdx.x;
    if (t >= ne * FD) return;
    int e = t >> 6, k = t & 63;
    atomicAdd(&out[(size_t)recv[e] * FD + k], last[(size_t)src[e] * FD + k]);
}

__global__ void scale_kernel(float* __restrict__ buf,
                             const float* __restrict__ degs, int idx, int n) {
    int t = blockIdx.x * blockDim.x + threadIdx.x;
    if (t < n) buf[t] = buf[t] / degs[idx];
}

// ---------------------------------------------------------------------------
// Dropout (deterministic hash-based Bernoulli, p_keep = 0.6, scale 1/0.6)
// ---------------------------------------------------------------------------
__global__ void dropout_kernel(const float* __restrict__ in,
                               float* __restrict__ out, int n) {
    int t = blockIdx.x * blockDim.x + threadIdx.x;
    if (t >= n) return;
    unsigned z = (unsigned)t * 0x9E3779B9u + 42u * 0x85EBCA6Bu;
    z ^= z >> 16; z *= 0x7FEB352Du;
    z ^= z >> 15; z *= 0x846CA68Bu;
    z ^= z >> 16;
    bool keep = (z >> 8) < (unsigned)(0.6 * 16777216.0);
    out[t] = keep ? in[t] * (1.0f / 0.6f) : 0.0f;
}

// ---------------------------------------------------------------------------
// Predictor second GEMM: Y[n,2] = relu(H[n,:]) @ W2[64,2]  (too skinny for WMMA)
// ---------------------------------------------------------------------------
__global__ void pred2_gemm_kernel(const float* __restrict__ H,
                                  const float* __restrict__ W2,
                                  float* __restrict__ Y, int n) {
    int t = blockIdx.x * blockDim.x + threadIdx.x;
    if (t >= n) return;
    const float* hr = H + (size_t)t * FD;
    float acc0 = 0.f, acc1 = 0.f;
    #pragma unroll
    for (int k = 0; k < FD; ++k) {
        float v = fmaxf(hr[k], 0.f);
        acc0 += v * W2[k * 2 + 0];
        acc1 += v * W2[k * 2 + 1];
    }
    Y[(size_t)t * 2 + 0] = acc0;
    Y[(size_t)t * 2 + 1] = acc1;
}

// ---------------------------------------------------------------------------
static inline unsigned gblocks(long long n, int b) {
    return (unsigned)((n + b - 1) / b);
}

extern "C" void kernel_launch(void* const* d_in, const int* in_sizes, int n_in,
                              void* d_out, int out_size, void* d_ws, size_t ws_size,
                              hipStream_t stream) {
    (void)in_sizes; (void)n_in; (void)out_size; (void)ws_size;

    const float* x    = (const float*)d_in[0];   // [S,N,64]
    const int*   ei   = (const int*)  d_in[1];   // [S,2,E]
    const int*   lei  = (const int*)  d_in[2];   // [S-1,2,EL]
    const float* degs = (const float*)d_in[3];   // [S-1]
    const float* W1   = (const float*)d_in[4];   // [S,64,64]
    const float* b1   = (const float*)d_in[5];   // [S,64]
    const float* W2   = (const float*)d_in[6];   // [S,64,64]
    const float* b2   = (const float*)d_in[7];   // [S,64]
    const float* pW1  = (const float*)d_in[8];   // [64,64]
    const float* pb1  = (const float*)d_in[9];   // [64]
    const float* pW2  = (const float*)d_in[10];  // [64,2]
    const float* pb2  = (const float*)d_in[11];  // [2]
    float*       out  = (float*)d_out;           // [N,2]

    const size_t NF = (size_t)N_NODES * FD;
    float* ws    = (float*)d_ws;
    float* bufA  = ws;                // [N,64] GEMM scratch
    float* bufB  = bufA + NF;         // [N,64] conv1 output
    float* bufL0 = bufB + NF;         // [N,64] surface/last rotate 0
    float* bufL1 = bufL0 + NF;        // [N,64] surface/last rotate 1
    float* deg   = bufL1 + NF;        // [N]
    float* dinv  = deg + N_NODES;     // [N]

    const int B = 256;
    const unsigned gN    = gblocks(N_NODES, B);
    const unsigned gE    = gblocks(NE, B);
    const unsigned gNF   = gblocks((long long)N_NODES * FD, B);
    const unsigned gEF   = gblocks((long long)NE * FD, B);
    const unsigned gELF  = gblocks((long long)NEL * FD, B);
    const unsigned gGemm = gblocks((N_NODES + 31) / 32, 8);  // 8 waves/block, 32 rows/wave

    float* last = nullptr;

    for (int i = 0; i < NSURF; ++i) {
        const int*   src = ei + (size_t)i * 2 * NE;
        const int*   dst = src + NE;
        const float* xi  = x  + (size_t)i * NF;

        // per-surface degree normalization (self-loops give deg >= 1)
        fill_kernel<<<gN, B, 0, stream>>>(deg, 1.0f, N_NODES);
        deg_edges_kernel<<<gE, B, 0, stream>>>(dst, deg, NE);
        dinv_kernel<<<gN, B, 0, stream>>>(deg, dinv, N_NODES);

        // conv1: bufA = x @ W1 ; bufB = aggregate(bufA) + b1
        gemm64_wmma<0><<<gGemm, B, 0, stream>>>(xi, W1 + (size_t)i * FD * FD, bufA, N_NODES);
        agg_self_kernel<FD><<<gNF, B, 0, stream>>>(bufA, dinv, b1 + (size_t)i * FD, bufB, N_NODES);
        agg_edges_kernel<FD><<<gEF, B, 0, stream>>>(bufA, src, dst, dinv, bufB, NE);

        // conv2: bufA = relu(bufB) @ W2 ; lf = aggregate(bufA) + b2
        gemm64_wmma<1><<<gGemm, B, 0, stream>>>(bufB, W2 + (size_t)i * FD * FD, bufA, N_NODES);
        float* lf   = (i & 1) ? bufL1 : bufL0;
        float* prev = (i & 1) ? bufL0 : bufL1;
        agg_self_kernel<FD><<<gNF, B, 0, stream>>>(bufA, dinv, b2 + (size_t)i * FD, lf, N_NODES);
        agg_edges_kernel<FD><<<gEF, B, 0, stream>>>(bufA, src, dst, dinv, lf, NE);

        if (i > 0) {
            const int* recv = lei + (size_t)(i - 1) * 2 * NEL;
            const int* lsrc = recv + NEL;
            cross_edges_kernel<<<gELF, B, 0, stream>>>(prev, recv, lsrc, lf, NEL);
            scale_kernel<<<gNF, B, 0, stream>>>(lf, degs, i - 1, (int)((long long)N_NODES * FD));
        }
        last = lf;
    }

    // dropout -> x_emb (bufA)
    dropout_kernel<<<gNF, B, 0, stream>>>(last, bufA, (int)((long long)N_NODES * FD));

    // predictor conv1 (reuses surface-2 dinv; edge_index[-1] == edge_index[2])
    const int* src2 = ei + (size_t)(NSURF - 1) * 2 * NE;
    const int* dst2 = src2 + NE;
    gemm64_wmma<0><<<gGemm, B, 0, stream>>>(bufA, pW1, bufB, N_NODES);
    agg_self_kernel<FD><<<gNF, B, 0, stream>>>(bufB, dinv, pb1, bufA, N_NODES);
    agg_edges_kernel<FD><<<gEF, B, 0, stream>>>(bufB, src2, dst2, dinv, bufA, NE);

    // predictor conv2: bufB[:N*2] = relu(bufA) @ pW2 ; out = aggregate + pb2
    pred2_gemm_kernel<<<gN, B, 0, stream>>>(bufA, pW2, bufB, N_NODES);
    agg_self_kernel<2><<<gblocks((long long)N_NODES * 2, B), B, 0, stream>>>(bufB, dinv, pb2, out, N_NODES);
    agg_edges_kernel<2><<<gblocks((long long)NE * 2, B), B, 0, stream>>>(bufB, src2, dst2, dinv, out, NE);
}